// GNN_63651415326879
// MI455X (gfx1250) — compile-verified
//
#include <hip/hip_runtime.h>
#include <hip/hip_bf16.h>
#include <math.h>

// ---------------------------------------------------------------------------
// GNN forward for MI455X (gfx1250, wave32).
// fp32 end-to-end using V_WMMA_F32_16X16X4_F32 (one 16-row tile per wave).
// BN params pre-folded once; waves grid-stride over tiles so per-wave stage
// registers (weights in B-fragment layout + folded BN) are amortized.
// ---------------------------------------------------------------------------

typedef __attribute__((ext_vector_type(2))) float v2f;
typedef __attribute__((ext_vector_type(8))) float v8f;

constexpr int   BB      = 32;
constexpr int   NN      = 10000;
constexpr int   EE      = 160000;
constexpr float EPSBN   = 1e-3f;
constexpr float EPSL2   = 1e-12f;
#define LSTRIDE 36   // floats per LDS tile row (16B-aligned rows, padded)

struct FfnParams {
  const float *g1,*b1,*m1,*v1,*W1,*bb1;   // stage 0: BN(d_in), W (d_in x 16), b(16)
  const float *g2,*b2,*m2,*v2,*W2,*bb2;   // stage 1: BN(16),  W (16 x 16),  b(16)
};
struct AllFfn { FfnParams f[6]; int k1[6]; };   // k1 = stage-0 input dim (16 or 32)

// Per-wave register-resident stage: weights in B-fragment layout, BN folded.
struct Stage16 { v2f bw[4]; float bias; float mul[8];  float add[8];  };
struct Stage32 { v2f bw[8]; float bias; float mul[16]; float add[16]; };

__device__ __forceinline__ void lds_fence() {
  // DS ops from one wave execute in order at the LDS; this wait + compiler
  // barrier makes the cross-lane tile handoff explicit.
  asm volatile("s_wait_dscnt 0x0" ::: "memory");
}

__device__ __forceinline__ float gelu_exact(float x) {
  return 0.5f * x * (1.0f + erff(x * 0.70710678118654752440f));
}

// ---- stage loaders: B-fragment v[j] holds W[k][n], k = 4c + 2h + j, n = lane&15
// Fm/Fa point at the pre-folded BN (mul/add) tables in workspace.
__device__ __forceinline__ void load_stage16(Stage16& S,
    const float* Fm, const float* Fa, const float* W, const float* bi, int lane)
{
  const int h = lane >> 4, n = lane & 15;
  S.bias = bi[n];
#pragma unroll
  for (int c = 0; c < 4; ++c) {
#pragma unroll
    for (int j = 0; j < 2; ++j) {
      const int k = 4*c + 2*h + j;
      S.bw[c][j]   = W[k*16 + n];
      S.mul[2*c+j] = Fm[k];
      S.add[2*c+j] = Fa[k];
    }
  }
}

__device__ __forceinline__ void load_stage32(Stage32& S,
    const float* Fm, const float* Fa, const float* W, const float* bi, int lane)
{
  const int h = lane >> 4, n = lane & 15;
  S.bias = bi[n];
#pragma unroll
  for (int c = 0; c < 8; ++c) {
#pragma unroll
    for (int j = 0; j < 2; ++j) {
      const int k = 4*c + 2*h + j;
      S.bw[c][j]   = W[k*16 + n];
      S.mul[2*c+j] = Fm[k];
      S.add[2*c+j] = Fa[k];
    }
  }
}

// ---- apply one stage: A frag from LDS tile (lane = row, slot = K), BN folded in.
__device__ __forceinline__ v8f apply_stage16(const Stage16& S, const float* T, int lane)
{
  const int h = lane >> 4, r = lane & 15;
  v8f acc;
#pragma unroll
  for (int j = 0; j < 8; ++j) acc[j] = S.bias;
#pragma unroll
  for (int c = 0; c < 4; ++c) {
    v2f a;
#pragma unroll
    for (int j = 0; j < 2; ++j) {
      const int k = 4*c + 2*h + j;
      a[j] = T[r*LSTRIDE + k] * S.mul[2*c+j] + S.add[2*c+j];
    }
    acc = __builtin_amdgcn_wmma_f32_16x16x4_f32(false, a, false, S.bw[c],
                                                (short)0, acc, false, false);
  }
  return acc;
}

__device__ __forceinline__ v8f apply_stage32(const Stage32& S, const float* T, int lane)
{
  const int h = lane >> 4, r = lane & 15;
  v8f acc;
#pragma unroll
  for (int j = 0; j < 8; ++j) acc[j] = S.bias;
#pragma unroll
  for (int c = 0; c < 8; ++c) {
    v2f a;
#pragma unroll
    for (int j = 0; j < 2; ++j) {
      const int k = 4*c + 2*h + j;
      a[j] = T[r*LSTRIDE + k] * S.mul[2*c+j] + S.add[2*c+j];
    }
    acc = __builtin_amdgcn_wmma_f32_16x16x4_f32(false, a, false, S.bw[c],
                                                (short)0, acc, false, false);
  }
  return acc;
}

// D frag: lane holds rows (j + 8h), column n = lane&15. GELU + store to tile.
__device__ __forceinline__ void gelu_store(const v8f& d, float* T, int lane)
{
  const int h = lane >> 4, n = lane & 15;
#pragma unroll
  for (int j = 0; j < 8; ++j) {
    const int row = j + 8*h;
    T[row*LSTRIDE + n] = gelu_exact(d[j]);
  }
}

// ---------------------------------------------------------------------------
// Setup: fold BN once.  fold layout per FFN (128 floats):
//   [0..31] mul1  [32..63] add1  [64..79] mul2  [80..95] add2
// FFN index: 0=pre 1=c1_msg 2=c1_upd 3=c2_msg 4=c2_upd 5=post
// ---------------------------------------------------------------------------
__global__ void k_fold(AllFfn A, float* __restrict__ fold)
{
  const int i = blockIdx.x;
  const FfnParams P = A.f[i];
  float* F = fold + i*128;
  const int t = threadIdx.x;
  if (t < A.k1[i]) {
    const float m = P.g1[t] * rsqrtf(P.v1[t] + EPSBN);
    F[t]      = m;
    F[32 + t] = P.b1[t] - P.m1[t]*m;
  }
  if (t < 16) {
    const float m = P.g2[t] * rsqrtf(P.v2[t] + EPSBN);
    F[64 + t] = m;
    F[80 + t] = P.b2[t] - P.m2[t]*m;
  }
}

// ---------------------------------------------------------------------------
// Main kernels: one 16-row tile per wave, grid-stride over tiles,
// 8 waves / 256-thread block.
// ---------------------------------------------------------------------------

__global__ void __launch_bounds__(256)
k_pre(const float* __restrict__ ni, FfnParams P, const float* __restrict__ F,
      float* __restrict__ x)
{
  __shared__ float Tl[8][16*LSTRIDE];
  const int lane = threadIdx.x & 31, wave = threadIdx.x >> 5;
  float* T = Tl[wave];
  const int r = lane & 15, h = lane >> 4;
  const int tilesPerB = NN / 16;
  const int total = BB * tilesPerB;
  const int nWaves = gridDim.x * 8;

  Stage32 S1; load_stage32(S1, F,      F+32, P.W1, P.bb1, lane);
  Stage16 S2; load_stage16(S2, F+64,   F+80, P.W2, P.bb2, lane);

  for (int wid = blockIdx.x*8 + wave; wid < total; wid += nWaves) {
    const int b = wid / tilesPerB, tile = wid % tilesPerB;
    { // load 16x32 input tile: lane loads 16 floats of its row half
      const float* src = ni + ((size_t)b*NN + (size_t)tile*16 + r) * 32 + 16*h;
#pragma unroll
      for (int i = 0; i < 4; ++i)
        *(float4*)&T[r*LSTRIDE + 16*h + 4*i] = ((const float4*)src)[i];
    }
    lds_fence();
    v8f d1 = apply_stage32(S1, T, lane);
    gelu_store(d1, T, lane);
    lds_fence();
    v8f d2 = apply_stage16(S2, T, lane);
    gelu_store(d2, T, lane);
    lds_fence();
    { // coalesced writeout
      float* dst = x + ((size_t)b*NN + (size_t)tile*16 + r) * 16 + 8*h;
      ((float4*)dst)[0] = *(const float4*)&T[r*LSTRIDE + 8*h];
      ((float4*)dst)[1] = *(const float4*)&T[r*LSTRIDE + 8*h + 4];
    }
  }
}

__global__ void __launch_bounds__(256)
k_msg(const float* __restrict__ x, float* __restrict__ agg,
      const int* __restrict__ nodei, const int* __restrict__ neigh,
      const float* __restrict__ bwts, FfnParams P, const float* __restrict__ F)
{
  __shared__ float Tl[8][16*LSTRIDE];
  const int lane = threadIdx.x & 31, wave = threadIdx.x >> 5;
  float* T = Tl[wave];
  const int r = lane & 15, h = lane >> 4;
  const int tilesPerB = EE / 16;
  const int total = BB * tilesPerB;
  const int nWaves = gridDim.x * 8;

  Stage16 S1; load_stage16(S1, F,    F+32, P.W1, P.bb1, lane);
  Stage16 S2; load_stage16(S2, F+64, F+80, P.W2, P.bb2, lane);

  for (int wid = blockIdx.x*8 + wave; wid < total; wid += nWaves) {
    const int b = wid / tilesPerB, tile = wid % tilesPerB;
    const int e  = tile*16 + r;
    const int nb = neigh[e];
    { // gather neighbour feature row (x is 20.5 MB -> L2 resident)
      const float* src = x + ((size_t)b*NN + nb) * 16 + 8*h;
      *(float4*)&T[r*LSTRIDE + 8*h]     = ((const float4*)src)[0];
      *(float4*)&T[r*LSTRIDE + 8*h + 4] = ((const float4*)src)[1];
    }
    lds_fence();
    v8f d1 = apply_stage16(S1, T, lane);
    gelu_store(d1, T, lane);
    lds_fence();
    v8f d2 = apply_stage16(S2, T, lane);
    gelu_store(d2, T, lane);
    lds_fence();
    { // scale by branch weight and segment-sum scatter (f32 atomics hit L2)
      const float w  = bwts[(size_t)b*EE + e];
      const int   dn = nodei[e];
      float* out = agg + ((size_t)b*NN + dn) * 16 + 8*h;
#pragma unroll
      for (int i = 0; i < 8; ++i)
        atomicAdd(&out[i], T[r*LSTRIDE + 8*h + i] * w);
    }
  }
}

__global__ void __launch_bounds__(256)
k_upd(float* __restrict__ x, const float* __restrict__ agg, FfnParams P,
      const float* __restrict__ F)
{
  __shared__ float Tl[8][16*LSTRIDE];
  const int lane = threadIdx.x & 31, wave = threadIdx.x >> 5;
  float* T = Tl[wave];
  const int r = lane & 15, h = lane >> 4;
  const int tilesPerB = NN / 16;
  const int total = BB * tilesPerB;
  const int nWaves = gridDim.x * 8;

  Stage32 S1; load_stage32(S1, F,    F+32, P.W1, P.bb1, lane);
  Stage16 S2; load_stage16(S2, F+64, F+80, P.W2, P.bb2, lane);

  for (int wid = blockIdx.x*8 + wave; wid < total; wid += nWaves) {
    const int b = wid / tilesPerB, tile = wid % tilesPerB;
    const size_t rowbase = ((size_t)b*NN + (size_t)tile*16 + r) * 16;
    { // h = concat([x, agg]) -> tile cols 0..15 | 16..31
      const float* sx = x   + rowbase + 8*h;
      const float* sa = agg + rowbase + 8*h;
      *(float4*)&T[r*LSTRIDE + 8*h]          = ((const float4*)sx)[0];
      *(float4*)&T[r*LSTRIDE + 8*h + 4]      = ((const float4*)sx)[1];
      *(float4*)&T[r*LSTRIDE + 16 + 8*h]     = ((const float4*)sa)[0];
      *(float4*)&T[r*LSTRIDE + 16 + 8*h + 4] = ((const float4*)sa)[1];
    }
    lds_fence();
    v8f d1 = apply_stage32(S1, T, lane);
    gelu_store(d1, T, lane);
    lds_fence();
    v8f d2 = apply_stage16(S2, T, lane);
    gelu_store(d2, T, lane);
    lds_fence();
    { // L2-normalize each row (cross-half combine via shfl_xor 16), + residual
      float v[8], ssq = 0.0f;
#pragma unroll
      for (int i = 0; i < 8; ++i) { v[i] = T[r*LSTRIDE + 8*h + i]; ssq += v[i]*v[i]; }
      ssq += __shfl_xor(ssq, 16);
      const float rs = rsqrtf(fmaxf(ssq, EPSL2));
      float* xr = x + rowbase + 8*h;
#pragma unroll
      for (int i = 0; i < 8; ++i) xr[i] = v[i]*rs + xr[i];
    }
  }
}

__global__ void __launch_bounds__(256)
k_post(const float* __restrict__ x, float* __restrict__ logits, FfnParams P,
       const float* __restrict__ F, const float* __restrict__ lw,
       const float* __restrict__ lb)
{
  __shared__ float Tl[8][16*LSTRIDE];
  const int lane = threadIdx.x & 31, wave = threadIdx.x >> 5;
  float* T = Tl[wave];
  const int r = lane & 15, h = lane >> 4;
  const int tilesPerB = NN / 16;
  const int total = BB * tilesPerB;
  const int nWaves = gridDim.x * 8;

  Stage16 S1; load_stage16(S1, F,    F+32, P.W1, P.bb1, lane);
  Stage16 S2; load_stage16(S2, F+64, F+80, P.W2, P.bb2, lane);
  const float lwk = lw[8*h + 0];  // dummy touch avoided; real loads below
  (void)lwk;

  for (int wid = blockIdx.x*8 + wave; wid < total; wid += nWaves) {
    const int b = wid / tilesPerB, tile = wid % tilesPerB;
    {
      const float* src = x + ((size_t)b*NN + (size_t)tile*16 + r) * 16 + 8*h;
      *(float4*)&T[r*LSTRIDE + 8*h]     = ((const float4*)src)[0];
      *(float4*)&T[r*LSTRIDE + 8*h + 4] = ((const float4*)src)[1];
    }
    lds_fence();
    v8f d1 = apply_stage16(S1, T, lane);
    gelu_store(d1, T, lane);
    lds_fence();
    v8f d2 = apply_stage16(S2, T, lane);
    gelu_store(d2, T, lane);
    lds_fence();
    { // logits = row . logits_W + logits_b
      float p = 0.0f;
#pragma unroll
      for (int i = 0; i < 8; ++i) p += T[r*LSTRIDE + 8*h + i] * lw[8*h + i];
      p += __shfl_xor(p, 16);
      if (h == 0) logits[(size_t)b*NN + (size_t)tile*16 + r] = p + lb[0];
    }
  }
}

__global__ void __launch_bounds__(256)
k_critic(const float* __restrict__ logits, const float* __restrict__ cw,
         const float* __restrict__ cb, float* __restrict__ out)
{
  __shared__ float red[256];
  const int b = blockIdx.x;
  float acc = 0.0f;
  for (int n = threadIdx.x; n < NN; n += 256)
    acc += logits[(size_t)b*NN + n] * cw[n];
  red[threadIdx.x] = acc;
  __syncthreads();
  for (int s = 128; s > 0; s >>= 1) {
    if (threadIdx.x < (unsigned)s) red[threadIdx.x] += red[threadIdx.x + s];
    __syncthreads();
  }
  if (threadIdx.x == 0) out[b] = red[0] + cb[0];
}

__global__ void k_zero(float* __restrict__ p, size_t n)
{
  size_t i = (size_t)blockIdx.x * blockDim.x + threadIdx.x;
  const size_t stride = (size_t)gridDim.x * blockDim.x;
  for (; i < n; i += stride) p[i] = 0.0f;
}

// ---------------------------------------------------------------------------
// Host launcher. Input order: node_info, branch_weights, node_indices,
// neighbour_indices, then params flattened as a pytree with dict keys sorted:
// c1_msg[4..15] c1_upd[16..27] c2_msg[28..39] c2_upd[40..51]
// critic_W[52] critic_b[53] logits_W[54] logits_b[55] post[56..67] pre[68..79]
// (each stage tuple: gamma, beta, mean, var, W, b)
// ---------------------------------------------------------------------------
static FfnParams ffn_at(void* const* d_in, int base)
{
  FfnParams f;
  f.g1  = (const float*)d_in[base+0];  f.b1  = (const float*)d_in[base+1];
  f.m1  = (const float*)d_in[base+2];  f.v1  = (const float*)d_in[base+3];
  f.W1  = (const float*)d_in[base+4];  f.bb1 = (const float*)d_in[base+5];
  f.g2  = (const float*)d_in[base+6];  f.b2  = (const float*)d_in[base+7];
  f.m2  = (const float*)d_in[base+8];  f.v2  = (const float*)d_in[base+9];
  f.W2  = (const float*)d_in[base+10]; f.bb2 = (const float*)d_in[base+11];
  return f;
}

extern "C" void kernel_launch(void* const* d_in, const int* in_sizes, int n_in,
                              void* d_out, int out_size, void* d_ws, size_t ws_size,
                              hipStream_t stream)
{
  const float* node_info = (const float*)d_in[0];
  const float* bwts      = (const float*)d_in[1];
  const int*   nodei     = (const int*)d_in[2];
  const int*   neigh     = (const int*)d_in[3];

  const FfnParams c1_msg = ffn_at(d_in, 4);
  const FfnParams c1_upd = ffn_at(d_in, 16);
  const FfnParams c2_msg = ffn_at(d_in, 28);
  const FfnParams c2_upd = ffn_at(d_in, 40);
  const float* critic_W  = (const float*)d_in[52];
  const float* critic_b  = (const float*)d_in[53];
  const float* logits_W  = (const float*)d_in[54];
  const float* logits_b  = (const float*)d_in[55];
  const FfnParams post   = ffn_at(d_in, 56);
  const FfnParams pre    = ffn_at(d_in, 68);

  const size_t featElems = (size_t)BB * NN * 16;   // 5.12M floats
  float* xbuf   = (float*)d_ws;
  float* agg    = xbuf + featElems;
  float* logits = agg + featElems;
  float* fold   = logits + (size_t)BB * NN;        // 6*128 floats of folded BN

  // fold BN once: index 0=pre 1=c1_msg 2=c1_upd 3=c2_msg 4=c2_upd 5=post
  AllFfn A;
  A.f[0] = pre;    A.k1[0] = 32;
  A.f[1] = c1_msg; A.k1[1] = 16;
  A.f[2] = c1_upd; A.k1[2] = 32;
  A.f[3] = c2_msg; A.k1[3] = 16;
  A.f[4] = c2_upd; A.k1[4] = 32;
  A.f[5] = post;   A.k1[5] = 16;
  k_fold<<<6, 32, 0, stream>>>(A, fold);

  const dim3 blk(256);
  const int nodeBlocks = 1024;   // 8192 waves over 20000 node tiles
  const int edgeBlocks = 4096;   // 32768 waves over 320000 edge tiles

  k_pre<<<nodeBlocks, blk, 0, stream>>>(node_info, pre, fold + 0*128, xbuf);

  // conv 1
  k_zero<<<2048, blk, 0, stream>>>(agg, featElems);
  k_msg<<<edgeBlocks, blk, 0, stream>>>(xbuf, agg, nodei, neigh, bwts,
                                        c1_msg, fold + 1*128);
  k_upd<<<nodeBlocks, blk, 0, stream>>>(xbuf, agg, c1_upd, fold + 2*128);

  // conv 2
  k_zero<<<2048, blk, 0, stream>>>(agg, featElems);
  k_msg<<<edgeBlocks, blk, 0, stream>>>(xbuf, agg, nodei, neigh, bwts,
                                        c2_msg, fold + 3*128);
  k_upd<<<nodeBlocks, blk, 0, stream>>>(xbuf, agg, c2_upd, fold + 4*128);

  k_post<<<nodeBlocks, blk, 0, stream>>>(xbuf, logits, post, fold + 5*128,
                                         logits_W, logits_b);
  k_critic<<<BB, blk, 0, stream>>>(logits, critic_W, critic_b, (float*)d_out);
}